// GATLayer_33741263078269
// MI455X (gfx1250) — compile-verified
//
#include <hip/hip_runtime.h>
#include <math.h>

// ---------------------------------------------------------------------------
// GAT layer for MI455X (gfx1250, wave32).
// Phase 1: per-edge attention score via V_WMMA_F32_16X16X4_F32 (16 edges/wave)
// Phase 2: wave-per-node masked softmax + coalesced gather-weighted-sum.
// Memory-bound (~0.95 GB total traffic -> ~41 us @ 23.3 TB/s); f32 throughout.
// ---------------------------------------------------------------------------

typedef __attribute__((ext_vector_type(2))) float v2f;
typedef __attribute__((ext_vector_type(4))) float v4f;
typedef __attribute__((ext_vector_type(8))) float v8f;

__device__ __forceinline__ float softplus_f32(float x) {
  // stable: max(x,0) + log1p(exp(-|x|))
  return fmaxf(x, 0.0f) + log1pf(__expf(-fabsf(x)));
}

// ---------------------------------------------------------------------------
// Kernel 1: e[i] = softplus( dot(z_feature[i], w[0:128]) + dot(z_others[i], w[128:256]) )
// One wave computes 16 edges with 64 x V_WMMA_F32_16X16X4_F32.
//   A (16x4 f32): lane L -> edge m = L&15 ; half h = L>>4 holds K = 2h, 2h+1
//                 in its two A VGPRs  => v2f load at row + c + 2h.
//   B (4x16 f32): w chunk broadcast across all 16 N-columns, same per-lane
//                 K placement as A    => v2f load at w + c + 2h.
//   D (16x16): every column identical; lanes with (L&15)==0 hold M = 8*(L>>4)+r
//              in acc[r] and store the 16 scores.
// ---------------------------------------------------------------------------
__global__ __launch_bounds__(256) void gat_edge_score_wmma(
    const float* __restrict__ zf, const float* __restrict__ zo,
    const float* __restrict__ w, float* __restrict__ e, int E)
{
  const int lane  = threadIdx.x & 31;
  const long wave = (long)blockIdx.x * 8 + (threadIdx.x >> 5);
  const long base = wave * 16;
  if (base >= E) return;                       // whole wave in/out: EXEC all-1s

  const int  m     = lane & 15;                // edge within 16-group
  const int  khalf = (lane >> 4) << 1;         // 0 or 2 -> K placement
  long edge = base + m;
  if (edge >= E) edge = (long)E - 1;           // clamp (dup compute, store-guarded)

  const float* rowf = zf + edge * 128 + khalf;
  const float* rowo = zo + edge * 128 + khalf;
  const float* wf   = w + khalf;               // w[0:128]   for z_feature
  const float* wo   = w + 128 + khalf;         // w[128:256] for z_others

  v8f acc = {0.f, 0.f, 0.f, 0.f, 0.f, 0.f, 0.f, 0.f};

#pragma unroll 8
  for (int c = 0; c < 128; c += 4) {
    v2f a = *(const v2f*)(rowf + c);
    v2f b = *(const v2f*)(wf + c);
    acc = __builtin_amdgcn_wmma_f32_16x16x4_f32(false, a, false, b,
                                                (short)0, acc, false, false);
  }
#pragma unroll 8
  for (int c = 0; c < 128; c += 4) {
    v2f a = *(const v2f*)(rowo + c);
    v2f b = *(const v2f*)(wo + c);
    acc = __builtin_amdgcn_wmma_f32_16x16x4_f32(false, a, false, b,
                                                (short)0, acc, false, false);
  }

  if (m == 0) {                                // lanes 0 (M=0..7) and 16 (M=8..15)
    const long off = base + (long)(lane >> 4) * 8;
#pragma unroll
    for (int r = 0; r < 8; ++r) {
      const long idx = off + r;
      if (idx < E) e[idx] = softplus_f32(acc[r]);
    }
  }
}

// ---------------------------------------------------------------------------
// Kernel 2: per-node masked softmax over K=16 gathered scores, then
// f_up[n] = sum_k alpha[k] * z_others[scope[n][k]-1].
// One wave per OUTPUT row (row 0 = zeros). Lanes 0..15 hold one neighbor each
// (mirrored in 16..31 so width-16 xor reductions are uniform); the weighted
// sum reads one fully-coalesced 512B row per k (float4 per lane).
// ---------------------------------------------------------------------------
__global__ __launch_bounds__(256) void gat_node_aggregate(
    const float* __restrict__ zo, const int* __restrict__ scope,
    const float* __restrict__ e, float* __restrict__ out, int N)
{
  const int  lane = threadIdx.x & 31;
  const long row  = (long)blockIdx.x * 8 + (threadIdx.x >> 5);  // 0..N
  if (row > N) return;

  v4f acc = {0.f, 0.f, 0.f, 0.f};

  if (row > 0) {
    const long node = row - 1;
    const int  k    = lane & 15;

    const int   s    = scope[node * 16 + k];           // 0 == padding
    const float ek   = (s > 0) ? e[s - 1] : 0.0f;      // e_pad[s]
    const bool  mask = (ek != 0.0f);                   // softplus>0 => 0 only pad

    // masked max over the 16 slots (duplicated across wave halves)
    float mx = mask ? ek : -1e30f;
    for (int off = 8; off >= 1; off >>= 1)
      mx = fmaxf(mx, __shfl_xor(mx, off, 16));

    float p = mask ? __expf(ek - mx) : 0.0f;
    float sum = p;
    for (int off = 8; off >= 1; off >>= 1)
      sum += __shfl_xor(sum, off, 16);

    const float alpha = (sum > 0.0f) ? (p / sum) : 0.0f;

    // weighted neighbor sum: 16 coalesced 512B row gathers
#pragma unroll 4
    for (int kk = 0; kk < 16; ++kk) {
      const int   sk = __shfl(s, kk, 32);      // wave-uniform
      const float ak = __shfl(alpha, kk, 32);  // wave-uniform
      if (sk > 0 && ak != 0.0f) {
        const v4f zv = *(const v4f*)(zo + (long)(sk - 1) * 128 + lane * 4);
        acc += ak * zv;
      }
    }
  }

  *(v4f*)(out + row * 128 + lane * 4) = acc;
}

extern "C" void kernel_launch(void* const* d_in, const int* in_sizes, int n_in,
                              void* d_out, int out_size, void* d_ws, size_t ws_size,
                              hipStream_t stream) {
  const float* zf    = (const float*)d_in[0];   // [E,128]
  const float* zo    = (const float*)d_in[1];   // [E,128]
  const int*   scope = (const int*)d_in[2];     // [N,16]
  const float* w     = (const float*)d_in[3];   // [256]
  float*       out   = (float*)d_out;           // [N+1,128]
  float*       e     = (float*)d_ws;            // [E] edge scores (scratch)

  const int E = in_sizes[0] / 128;
  const int N = in_sizes[2] / 16;

  const dim3 blk(256);                          // 8 waves / block

  const int waves1 = (E + 15) / 16;             // 16 edges per wave
  gat_edge_score_wmma<<<dim3((waves1 + 7) / 8), blk, 0, stream>>>(zf, zo, w, e, E);

  const int rows = N + 1;                       // one wave per output row
  gat_node_aggregate<<<dim3((rows + 7) / 8), blk, 0, stream>>>(zo, scope, e, out, N);
}